// FusionAFTDymaicWModel_45664092291712
// MI455X (gfx1250) — compile-verified
//
#include <hip/hip_runtime.h>
#include <hip/hip_bf16.h>
#include <cstdint>

// =====================================================================
// FusionAFTDymaicWModel forward, MI455X (gfx1250, wave32, WMMA).
//
// KEY OPTIMIZATION (exact math, not approximation):
//   dyn_bias = aft_pos * cw[:,None,:] has seq-dim 1 and is added to k
//   before softmax over the seq axis (length 5). A constant shift per
//   (batch, feature) is softmax-invariant, so the whole
//   DynamicConditionNetwork (context MHA, ~950 GFLOP) is dead code and
//   is eliminated here. Remaining work ~190 GFLOP / ~1 GB traffic ->
//   memory bound at 23.3 TB/s.
//
// GEMMs use V_WMMA_F32_16X16X4_F32 (true fp32 matrix op) to keep full
// reference precision; the workload is HBM-bound so fp32 WMMA rate is
// not the bottleneck.
//
// GEMM details:
//  * B tile kept in LDS in a K-paired layout so each B-fragment is one
//    aligned ds_load_b64 (no v_mov shuffles feeding the WMMAs).
//  * Double-buffered LDS with a single barrier per K-block: next
//    block's global loads are issued before computing the current one.
//
// ASSUMPTIONS:
//  * params dict flattens in insertion order -> d_in indices below.
//  * ws_size >= ~504 MB (126M floats), buffers reused where possible.
// =====================================================================

#define B_ELEMS 8192
#define D_MODEL 768

typedef float v2f __attribute__((ext_vector_type(2)));
typedef float v8f __attribute__((ext_vector_type(8)));

static __device__ __forceinline__ v8f v8zero() {
  v8f z;
#pragma unroll
  for (int i = 0; i < 8; ++i) z[i] = 0.0f;
  return z;
}

static __device__ __forceinline__ v8f wmma_f32_k4(v2f a, v2f b, v8f c) {
  // D = A(16x4,f32) * B(4x16,f32) + C(16x16,f32)
  return __builtin_amdgcn_wmma_f32_16x16x4_f32(
      /*neg_a=*/false, a, /*neg_b=*/false, b,
      /*c_mod=*/(short)0, c, /*reuse_a=*/false, /*reuse_b=*/false);
}

// ---------------------------------------------------------------------
// Block reduction helper (256 threads).
// ---------------------------------------------------------------------
static __device__ __forceinline__ float block_sum_256(float v, float* red) {
  const int t = threadIdx.x;
  red[t] = v;
  __syncthreads();
  for (int s = 128; s > 0; s >>= 1) {
    if (t < s) red[t] += red[t + s];
    __syncthreads();
  }
  float r = red[0];
  __syncthreads();
  return r;
}

// ---------------------------------------------------------------------
// Generic fp32 WMMA GEMM:  C[M,N] = A[M,K] @ W[K,N] + bias[N]
// Block tile 128x64, 256 threads (8 waves), BK=16, double-buffered LDS.
// Requires: M % 128 == 0, N % 64 == 0, K % 16 == 0 (true for all uses).
// Per ISA 7.12.2 fragment layouts:
//   A 16x4 : lane<16 -> M=lane, K={0,1}; lane>=16 -> M=lane-16, K={2,3}
//   B 4x16 : lane<16 -> N=lane, K rows {0,1}; lane>=16 -> K rows {2,3}
//   C 16x16: vgpr j -> M = j + (lane>=16 ? 8 : 0), N = lane%16
// LDS B layout is K-paired: (k,n) -> ((k>>1)*64 + n)*2 + (k&1), so a
// B-fragment (two consecutive K rows at one n) is one b64 load.
// ---------------------------------------------------------------------
__global__ __launch_bounds__(256) void gemm_bias_f32(
    const float* __restrict__ A, const float* __restrict__ W,
    const float* __restrict__ bias, float* __restrict__ C,
    int M, int N, int K) {
  (void)M;
  __shared__ float lA[2][128 * 16];
  __shared__ float lB[2][16 * 64];

  const int t    = threadIdx.x;
  const int lane = t & 31;
  const int wave = t >> 5;           // 0..7 -> row tile
  const int n0   = blockIdx.x * 64;
  const int m0   = blockIdx.y * 128;

  v8f acc[4];
#pragma unroll
  for (int i = 0; i < 4; ++i) acc[i] = v8zero();

  const int nlo = lane & 15;          // N (or M for A-frag) within tile
  const int kp  = (lane >> 4) << 1;   // K sub-offset {0,2}

  // Staging index precompute.
  const int a_row0 = t >> 2;                 // A chunk 0 row
  const int a_col  = (t & 3) << 2;           // A chunk col (float4)
  const int b_kp   = t >> 5;                 // B: K-row pair 0..7
  const int b_n    = (t & 31) << 1;          // B: n within tile (pairs)

  const float* Ag = A + (size_t)(m0 + a_row0) * K + a_col;      // + kk
  const float* Ag2 = Ag + (size_t)64 * K;                        // row + 64
  const float* Wg0 = W + (size_t)(b_kp * 2) * N + n0 + b_n;      // + kk*N
  const float* Wg1 = Wg0 + N;

  // ---- prologue: stage block 0 -------------------------------------
  float4 ra0 = *(const float4*)(Ag);
  float4 ra1 = *(const float4*)(Ag2);
  float2 rb0 = *(const float2*)(Wg0);
  float2 rb1 = *(const float2*)(Wg1);
  *(float4*)(&lA[0][a_row0 * 16 + a_col]) = ra0;
  *(float4*)(&lA[0][(a_row0 + 64) * 16 + a_col]) = ra1;
  {
    float4 bi = make_float4(rb0.x, rb1.x, rb0.y, rb1.y);
    *(float4*)(&lB[0][(b_kp * 64 + b_n) * 2]) = bi;
  }
  __syncthreads();

  int buf = 0;
  for (int kk = 0; kk < K; kk += 16) {
    const bool has_next = (kk + 16) < K;
    // Issue next block's global loads first (latency hidden by compute).
    if (has_next) {
      ra0 = *(const float4*)(Ag + kk + 16);
      ra1 = *(const float4*)(Ag2 + kk + 16);
      rb0 = *(const float2*)(Wg0 + (size_t)(kk + 16) * N);
      rb1 = *(const float2*)(Wg1 + (size_t)(kk + 16) * N);
    }

    // Compute current block from LDS.
    const float* cA = lA[buf];
    const float* cB = lB[buf];
#pragma unroll
    for (int ks = 0; ks < 16; ks += 4) {
      const v2f a = *(const v2f*)(cA + (wave * 16 + nlo) * 16 + ks + kp);
      const int bbase = ((ks + kp) >> 1) * 128;  // (pairIdx*64)*2
#pragma unroll
      for (int tc = 0; tc < 4; ++tc) {
        const v2f b = *(const v2f*)(cB + bbase + (tc * 16 + nlo) * 2);
        acc[tc] = wmma_f32_k4(a, b, acc[tc]);
      }
    }

    // Store next block into the alternate buffer.
    if (has_next) {
      float* nA = lA[buf ^ 1];
      float* nB = lB[buf ^ 1];
      *(float4*)(&nA[a_row0 * 16 + a_col]) = ra0;
      *(float4*)(&nA[(a_row0 + 64) * 16 + a_col]) = ra1;
      float4 bi = make_float4(rb0.x, rb1.x, rb0.y, rb1.y);
      *(float4*)(&nB[(b_kp * 64 + b_n) * 2]) = bi;
    }
    __syncthreads();
    buf ^= 1;
  }

  // Epilogue: bias add + store.
  const int mhalf = (lane >> 4) * 8;
#pragma unroll
  for (int tc = 0; tc < 4; ++tc) {
    const int col = n0 + tc * 16 + nlo;
    const float bv = bias[col];
#pragma unroll
    for (int j = 0; j < 8; ++j) {
      const int row = m0 + wave * 16 + mhalf + j;
      C[(size_t)row * N + col] = acc[tc][j] + bv;
    }
  }
}

// ---------------------------------------------------------------------
// Stage 1 fused: DyT(raw,13) -> linear 13->256 (W & Wr) -> LN -> SiLU
//                -> conv1d(k=3,pad=1) -> + residual. One block per row.
// ---------------------------------------------------------------------
__global__ __launch_bounds__(256) void stage1_kernel(
    const float* __restrict__ raw, const float* __restrict__ alpha,
    const float* __restrict__ dw, const float* __restrict__ db,
    const float* __restrict__ Wm, const float* __restrict__ bm,
    const float* __restrict__ g, const float* __restrict__ beta,
    const float* __restrict__ cw, const float* __restrict__ cb,
    const float* __restrict__ Wr, const float* __restrict__ br,
    float* __restrict__ out) {
  __shared__ float xd[16];
  __shared__ float red[256];
  __shared__ float sbuf[258];
  const int e = blockIdx.x;
  const int t = threadIdx.x;

  if (t < 13) xd[t] = tanhf(alpha[0] * raw[e * 13 + t]) * dw[t] + db[t];
  __syncthreads();

  float h = bm[t], r = br[t];
#pragma unroll
  for (int c = 0; c < 13; ++c) {
    const float x = xd[c];
    h += x * Wm[c * 256 + t];
    r += x * Wr[c * 256 + t];
  }
  const float m   = block_sum_256(h, red) * (1.0f / 256.0f);
  const float m2  = block_sum_256(h * h, red) * (1.0f / 256.0f);
  const float inv = rsqrtf(m2 - m * m + 1e-5f);
  const float y   = (h - m) * inv * g[t] + beta[t];
  const float s   = y / (1.0f + expf(-y));  // SiLU
  sbuf[t + 1] = s;
  if (t == 0) { sbuf[0] = 0.0f; sbuf[257] = 0.0f; }
  __syncthreads();
  out[(size_t)e * 256 + t] =
      cw[0] * sbuf[t] + cw[1] * sbuf[t + 1] + cw[2] * sbuf[t + 2] + cb[0] + r;
}

// ---------------------------------------------------------------------
// Stage epilogue (widths 512/768): LN -> SiLU -> conv3 -> + residual.
// ---------------------------------------------------------------------
__global__ __launch_bounds__(256) void stage_epi_kernel(
    const float* __restrict__ H, const float* __restrict__ R,
    const float* __restrict__ g, const float* __restrict__ beta,
    const float* __restrict__ cw, const float* __restrict__ cb,
    float* __restrict__ out, int width) {
  __shared__ float red[256];
  __shared__ float sbuf[770];  // width + 2, width <= 768
  const int e = blockIdx.x;
  const int t = threadIdx.x;
  const int nv = width >> 8;   // 2 or 3
  const float* h = H + (size_t)e * width;

  float hv[3];
  float s0 = 0.0f, s1 = 0.0f;
  for (int i = 0; i < nv; ++i) {
    const float x = h[t + (i << 8)];
    hv[i] = x;
    s0 += x;
    s1 += x * x;
  }
  const float m   = block_sum_256(s0, red) / (float)width;
  const float m2  = block_sum_256(s1, red) / (float)width;
  const float inv = rsqrtf(m2 - m * m + 1e-5f);
  for (int i = 0; i < nv; ++i) {
    const int j = t + (i << 8);
    const float y = (hv[i] - m) * inv * g[j] + beta[j];
    sbuf[j + 1] = y / (1.0f + expf(-y));
  }
  if (t == 0) { sbuf[0] = 0.0f; sbuf[width + 1] = 0.0f; }
  __syncthreads();
  const float w0 = cw[0], w1 = cw[1], w2 = cw[2], c0 = cb[0];
  const float* r = R + (size_t)e * width;
  float* o = out + (size_t)e * width;
  for (int i = 0; i < nv; ++i) {
    const int j = t + (i << 8);
    o[j] = w0 * sbuf[j] + w1 * sbuf[j + 1] + w2 * sbuf[j + 2] + c0 + r[j];
  }
}

// ---------------------------------------------------------------------
// LN + SiLU in place (widths 512/1024).
// ---------------------------------------------------------------------
__global__ __launch_bounds__(256) void lnsilu_kernel(
    float* __restrict__ X, const float* __restrict__ g,
    const float* __restrict__ beta, int width) {
  __shared__ float red[256];
  const int e = blockIdx.x;
  const int t = threadIdx.x;
  const int nv = width >> 8;  // up to 4
  float* x = X + (size_t)e * width;

  float hv[4];
  float s0 = 0.0f, s1 = 0.0f;
  for (int i = 0; i < nv; ++i) {
    const float v = x[t + (i << 8)];
    hv[i] = v;
    s0 += v;
    s1 += v * v;
  }
  const float m   = block_sum_256(s0, red) / (float)width;
  const float m2  = block_sum_256(s1, red) / (float)width;
  const float inv = rsqrtf(m2 - m * m + 1e-5f);
  for (int i = 0; i < nv; ++i) {
    const int j = t + (i << 8);
    const float y = (hv[i] - m) * inv * g[j] + beta[j];
    x[j] = y / (1.0f + expf(-y));
  }
}

// ---------------------------------------------------------------------
// Blood branch: DyT(blood,7) -> linear 7->768. One block per row.
// ---------------------------------------------------------------------
__global__ __launch_bounds__(256) void blood_kernel(
    const float* __restrict__ blood, const float* __restrict__ alpha,
    const float* __restrict__ dw, const float* __restrict__ db,
    const float* __restrict__ BW, const float* __restrict__ bb,
    float* __restrict__ out) {
  __shared__ float bd[8];
  const int e = blockIdx.x;
  const int t = threadIdx.x;
  if (t < 7) bd[t] = tanhf(alpha[0] * blood[e * 7 + t]) * dw[t] + db[t];
  __syncthreads();
  for (int j = t; j < D_MODEL; j += 256) {
    float acc = bb[j];
#pragma unroll
    for (int c = 0; c < 7; ++c) acc += bd[c] * BW[c * D_MODEL + j];
    out[(size_t)e * D_MODEL + j] = acc;
  }
}

// ---------------------------------------------------------------------
// Concatenate x = [text, raw_enc, blood_enc, basic_risk, blood_risk]
// into (B*5, 768).
// ---------------------------------------------------------------------
__global__ __launch_bounds__(256) void concat_kernel(
    const float* __restrict__ text, const float* __restrict__ raw_enc,
    const float* __restrict__ blood_enc, const float* __restrict__ basic_risk,
    const float* __restrict__ blood_risk, float* __restrict__ x) {
  const size_t idx = (size_t)blockIdx.x * 256 + threadIdx.x;  // < B*5*768
  const int d = (int)(idx % D_MODEL);
  const int s = (int)((idx / D_MODEL) % 5);
  const size_t e = idx / (5 * D_MODEL);
  const size_t src = e * D_MODEL + d;
  float v;
  switch (s) {
    case 0: v = text[src]; break;
    case 1: v = raw_enc[src]; break;
    case 2: v = blood_enc[src]; break;
    case 3: v = basic_risk[src]; break;
    default: v = blood_risk[src]; break;
  }
  x[idx] = v;
}

// ---------------------------------------------------------------------
// AFT combine: per (e,d): w_s = softmax_s(k[e,s,d])  (dyn_bias is a
// constant over s -> softmax-invariant, eliminated); x2 = w * v.
// x2 written in place over K (same flat layout, per-thread RAW safe).
// ---------------------------------------------------------------------
__global__ __launch_bounds__(256) void aft_kernel(
    const float* __restrict__ Kb, const float* __restrict__ Vb,
    float* __restrict__ X2) {
  const size_t idx = (size_t)blockIdx.x * 256 + threadIdx.x;  // < B*768
  const size_t e = idx / D_MODEL;
  const int d = (int)(idx % D_MODEL);
  const size_t base = e * (5 * D_MODEL) + d;

  float kv[5];
  float mx = -3.0e38f;
#pragma unroll
  for (int s = 0; s < 5; ++s) {
    kv[s] = Kb[base + (size_t)s * D_MODEL];
    mx = fmaxf(mx, kv[s]);
  }
  float sum = 0.0f;
#pragma unroll
  for (int s = 0; s < 5; ++s) {
    kv[s] = expf(kv[s] - mx);
    sum += kv[s];
  }
  const float inv = 1.0f / sum;
#pragma unroll
  for (int s = 0; s < 5; ++s) {
    X2[base + (size_t)s * D_MODEL] = kv[s] * inv * Vb[base + (size_t)s * D_MODEL];
  }
}

// ---------------------------------------------------------------------
// Output heads: pred = sigmoid(x@out_W + out_b) -> d_out[0:B]
//               importance = x@attn_W + attn_b  -> d_out[B : B+23B]
// One block (8 waves) per row; wave-level reductions via shuffles.
// ---------------------------------------------------------------------
__global__ __launch_bounds__(256) void heads_kernel(
    const float* __restrict__ X, const float* __restrict__ outW,
    const float* __restrict__ outb, const float* __restrict__ attnW,
    const float* __restrict__ attnb, float* __restrict__ out) {
  __shared__ float xr[512];
  const int e = blockIdx.x;
  const int t = threadIdx.x;
  const int lane = t & 31;
  const int wave = t >> 5;
  for (int j = t; j < 512; j += 256) xr[j] = X[(size_t)e * 512 + j];
  __syncthreads();

  for (int d = wave; d < 24; d += 8) {
    float s = 0.0f;
    if (d < 23) {
      for (int j = lane; j < 512; j += 32) s += xr[j] * attnW[j * 23 + d];
    } else {
      for (int j = lane; j < 512; j += 32) s += xr[j] * outW[j];
    }
#pragma unroll
    for (int off = 16; off > 0; off >>= 1) s += __shfl_down(s, off);
    if (lane == 0) {
      if (d < 23) {
        out[(size_t)B_ELEMS + (size_t)e * 23 + d] = s + attnb[d];
      } else {
        out[e] = 1.0f / (1.0f + expf(-(s + outb[0])));
      }
    }
  }
}

// =====================================================================
// Host orchestration.
// d_in layout (dict-insertion-order flatten):
//  0 basic(unused) 1 blood 2 text 3 raw 4 basic_risk 5 blood_risk
//  6 context(dead) 7 cond_Win 8 cond_bin 9 Wqkv 10 bqkv 11 Wo 12 bo
//  13 cond_Wout 14 cond_bout (7..14 all dead: softmax shift-invariance)
//  15 dyt_raw_a 16 dyt_raw_w 17 dyt_raw_b 18 dyt_blood_a 19 dyt_blood_w
//  20 dyt_blood_b
//  21..28 stage1 {W,b,g,beta,cw,cb,Wr,br}   29..36 stage2   37..44 stage3
//  45 blood_W 46 blood_b 47 aft_pos(dead) 48 aft_Wk 49 aft_bk
//  50 aft_Wv 51 aft_bv 52 fus_W1 53 fus_b1 54 fus_g1 55 fus_be1
//  56 fus_W2 57 fus_b2 58 fus_g2 59 fus_be2 60 out_W 61 out_b
//  62 attn_W 63 attn_b
// =====================================================================
extern "C" void kernel_launch(void* const* d_in, const int* in_sizes, int n_in,
                              void* d_out, int out_size, void* d_ws,
                              size_t ws_size, hipStream_t stream) {
  (void)in_sizes; (void)n_in; (void)out_size; (void)ws_size;
  auto f = [&](int i) { return (const float*)d_in[i]; };
  float* outp = (float*)d_out;

  const size_t B = B_ELEMS;
  float* ws = (float*)d_ws;
  size_t off = 0;
  auto alloc = [&](size_t n) { float* p = ws + off; off += n; return p; };

  float* s1        = alloc(B * 256);       // stage1 out
  float* hbuf      = alloc(B * 768);       // linear pre-LN (stage2 then stage3)
  float* rbuf      = alloc(B * 768);       // residual (stage2 then stage3)
  float* s2        = alloc(B * 512);       // stage2 out
  float* raw_enc   = alloc(B * 768);
  float* blood_enc = alloc(B * 768);
  float* xbuf      = alloc(B * 5 * 768);   // x; later reused as f1 (B,1024)
  float* kbuf      = alloc(B * 5 * 768);   // k; reused in place as x2 (B,3840)
  float* vbuf      = alloc(B * 5 * 768);   // v; later reused as f2 (B,512)
  float* f1 = xbuf;
  float* f2 = vbuf;
  // Total: ~126M floats (~504 MB) of d_ws.

  // ---- raw stage encoder -------------------------------------------
  stage1_kernel<<<B, 256, 0, stream>>>(f(3), f(15), f(16), f(17), f(21), f(22),
                                       f(23), f(24), f(25), f(26), f(27), f(28),
                                       s1);
  // stage2: 256 -> 512
  gemm_bias_f32<<<dim3(512 / 64, B / 128), 256, 0, stream>>>(
      s1, f(29), f(30), hbuf, B, 512, 256);
  gemm_bias_f32<<<dim3(512 / 64, B / 128), 256, 0, stream>>>(
      s1, f(35), f(36), rbuf, B, 512, 256);
  stage_epi_kernel<<<B, 256, 0, stream>>>(hbuf, rbuf, f(31), f(32), f(33),
                                          f(34), s2, 512);
  // stage3: 512 -> 768
  gemm_bias_f32<<<dim3(768 / 64, B / 128), 256, 0, stream>>>(
      s2, f(37), f(38), hbuf, B, 768, 512);
  gemm_bias_f32<<<dim3(768 / 64, B / 128), 256, 0, stream>>>(
      s2, f(43), f(44), rbuf, B, 768, 512);
  stage_epi_kernel<<<B, 256, 0, stream>>>(hbuf, rbuf, f(39), f(40), f(41),
                                          f(42), raw_enc, 768);

  // ---- blood branch -------------------------------------------------
  blood_kernel<<<B, 256, 0, stream>>>(f(1), f(18), f(19), f(20), f(45), f(46),
                                      blood_enc);

  // ---- AFT block (cond network eliminated: see header comment) ------
  concat_kernel<<<(B * 5 * 768) / 256, 256, 0, stream>>>(
      f(2), raw_enc, blood_enc, f(4), f(5), xbuf);
  gemm_bias_f32<<<dim3(768 / 64, (5 * B) / 128), 256, 0, stream>>>(
      xbuf, f(48), f(49), kbuf, 5 * B, 768, 768);
  gemm_bias_f32<<<dim3(768 / 64, (5 * B) / 128), 256, 0, stream>>>(
      xbuf, f(50), f(51), vbuf, 5 * B, 768, 768);
  aft_kernel<<<(B * 768) / 256, 256, 0, stream>>>(kbuf, vbuf, kbuf);

  // ---- fusion MLP head ---------------------------------------------
  gemm_bias_f32<<<dim3(1024 / 64, B / 128), 256, 0, stream>>>(
      kbuf /* x2 (B,3840) */, f(52), f(53), f1, B, 1024, 3840);
  lnsilu_kernel<<<B, 256, 0, stream>>>(f1, f(54), f(55), 1024);
  gemm_bias_f32<<<dim3(512 / 64, B / 128), 256, 0, stream>>>(
      f1, f(56), f(57), f2, B, 512, 1024);
  lnsilu_kernel<<<B, 256, 0, stream>>>(f2, f(58), f(59), 512);

  heads_kernel<<<B, 256, 0, stream>>>(f2, f(60), f(61), f(62), f(63), outp);
}